// FixedModel_50276887167210
// MI455X (gfx1250) — compile-verified
//
#include <hip/hip_runtime.h>
#include <hip/hip_bf16.h>

typedef __attribute__((ext_vector_type(2))) float v2f;
typedef __attribute__((ext_vector_type(4))) float v4f;
typedef __attribute__((ext_vector_type(8))) float v8f;
typedef __attribute__((ext_vector_type(4))) int   v4i;

// Custom "WMMA layout" for a 16-node group (wave) w:
//   logical (node m in 0..15, slot d in 0..15) lives at
//   off(w,m,d) = w*256 + lane*8 + r,  lane = d + (m>=8 ? 16 : 0), r = m&7
// This makes: per-lane 8 contiguous floats (2 x b128), and WMMA D reg r at
// lane l correspond exactly to node (r + (l>=16)*8), slot (l&15).
__device__ __forceinline__ int stor_off(int g, int d) {
    // g = global node id, d = slot
    return ((g >> 4) << 8) + ((d + (((g >> 3) & 1) << 4)) << 3) + (g & 7);
}

__global__ void prep_kernel(const float* __restrict__ fp,
                            const float* __restrict__ dem,
                            const int*   __restrict__ adj,
                            const int*   __restrict__ inidx,
                            const int*   __restrict__ num_nodes_p,
                            float* __restrict__ weights,   // WMMA layout
                            int*   __restrict__ gidx,      // A-matrix gather layout
                            float* __restrict__ flow0,     // WMMA layout (iter 1 result)
                            float* __restrict__ negdem,    // [wave][16] : -demand, index m
                            int total_nodes)
{
    int g = blockIdx.x * blockDim.x + threadIdx.x;
    if (g >= total_nodes) return;
    const int N = *num_nodes_p;

    // ---- masked softmax over the 16 slots of node g ----
    float v[16];
    float mx = -3.4e38f;
#pragma unroll
    for (int d = 0; d < 16; ++d) {
        float x = fp[g * 16 + d];
        if (adj[g * 16 + d] == N) x -= 1.0e7f;
        v[d] = x;
        mx = fmaxf(mx, x);
    }
    float s = 0.f;
#pragma unroll
    for (int d = 0; d < 16; ++d) { v[d] = expf(v[d] - mx); s += v[d]; }
    float inv = 1.f / s;

    int w = g >> 4, m = g & 15;
    float nd = -dem[g];
    negdem[w * 16 + m] = nd;             // m == (m>>3)*8 + (m&7): matches D rows
    float r0 = fmaxf(nd, 0.f);           // relu(0 - demand): iteration 1 inflow is 0

#pragma unroll
    for (int d = 0; d < 16; ++d) {
        float wt = v[d] * inv;
        int o = stor_off(g, d);
        weights[o] = wt;
        flow0[o]   = wt * r0;
    }

    // ---- fused gather indices, stored in the WMMA A-matrix load order ----
    // lane l, entry e (0..7): pair p=e>>1, j=e&1 -> (node=l&15, slot=p*4+j+(l>=16?2:0))
    // inverse: slot d -> p=d>>2, j=d&3; j<2 : lane=m,    e=2p+j
    //                                  j>=2: lane=m+16,  e=2p+j-2
#pragma unroll
    for (int d = 0; d < 16; ++d) {
        int p = d >> 2, j = d & 3;
        int lane = (j < 2) ? m : (m + 16);
        int e    = (j < 2) ? (2 * p + j) : (2 * p + j - 2);
        int bb = inidx[(g * 16 + d) * 3 + 0];
        int sr = inidx[(g * 16 + d) * 3 + 1];
        int sl = inidx[(g * 16 + d) * 3 + 2];
        int g2 = bb * N + sr;                       // global source node
        gidx[w * 256 + lane * 8 + e] = stor_off(g2, sl);
    }
}

__global__ void iter_kernel(const float* __restrict__ fin,      // WMMA layout
                            const float* __restrict__ weights,  // WMMA layout
                            const int*   __restrict__ gidx,
                            const float* __restrict__ negdem,
                            float* __restrict__ fout,
                            int store_std,                      // 1: standard [g][d] layout
                            int nwaves)
{
    int lane = threadIdx.x & 31;
    int w = (blockIdx.x * blockDim.x + threadIdx.x) >> 5;
    if (w >= nwaves) return;                 // wave-uniform: EXEC stays all-ones

    int lbase = w * 256 + lane * 8;

    // coalesced b128 index loads, then 8 scattered gathers (L2-resident)
    v4i g0 = *(const v4i*)(gidx + lbase);
    v4i g1 = *(const v4i*)(gidx + lbase + 4);
    float a0 = fin[g0.x], a1 = fin[g0.y], a2 = fin[g0.z], a3 = fin[g0.w];
    float a4 = fin[g1.x], a5 = fin[g1.y], a6 = fin[g1.z], a7 = fin[g1.w];

    // C init = -demand (broadcast within each half-wave): folds the subtraction
    int hi = (lane >> 4) << 3;
    v4f n0 = *(const v4f*)(negdem + w * 16 + hi);
    v4f n1 = *(const v4f*)(negdem + w * 16 + hi + 4);
    v8f c = { n0.x, n0.y, n0.z, n0.w, n1.x, n1.y, n1.z, n1.w };

    v2f ones = { 1.f, 1.f };                 // B = 4x16 all-ones => row sums
    v2f p0 = { a0, a1 }, p1 = { a2, a3 }, p2 = { a4, a5 }, p3 = { a6, a7 };
    // D[m,*] = sum_k flow[m,k] - demand[m], chained over K = 16 slots
    c = __builtin_amdgcn_wmma_f32_16x16x4_f32(false, p0, false, ones, (short)0, c, false, false);
    c = __builtin_amdgcn_wmma_f32_16x16x4_f32(false, p1, false, ones, (short)0, c, false, false);
    c = __builtin_amdgcn_wmma_f32_16x16x4_f32(false, p2, false, ones, (short)0, c, false, false);
    c = __builtin_amdgcn_wmma_f32_16x16x4_f32(false, p3, false, ones, (short)0, c, false, false);

    v4f w0 = *(const v4f*)(weights + lbase);
    v4f w1 = *(const v4f*)(weights + lbase + 4);

    v4f o0, o1;
    o0.x = w0.x * fmaxf(c[0], 0.f);
    o0.y = w0.y * fmaxf(c[1], 0.f);
    o0.z = w0.z * fmaxf(c[2], 0.f);
    o0.w = w0.w * fmaxf(c[3], 0.f);
    o1.x = w1.x * fmaxf(c[4], 0.f);
    o1.y = w1.y * fmaxf(c[5], 0.f);
    o1.z = w1.z * fmaxf(c[6], 0.f);
    o1.w = w1.w * fmaxf(c[7], 0.f);

    if (!store_std) {
        *(v4f*)(fout + lbase)     = o0;      // 2 x global_store_b128, coalesced
        *(v4f*)(fout + lbase + 4) = o1;
    } else {
        // final iteration: un-permute to standard [node][slot] layout
        int col = lane & 15;
        int gbase = (w * 16 + hi) * 16 + col;   // node = w*16 + hi + r
        fout[gbase + 0 * 16] = o0.x;
        fout[gbase + 1 * 16] = o0.y;
        fout[gbase + 2 * 16] = o0.z;
        fout[gbase + 3 * 16] = o0.w;
        fout[gbase + 4 * 16] = o1.x;
        fout[gbase + 5 * 16] = o1.y;
        fout[gbase + 6 * 16] = o1.z;
        fout[gbase + 7 * 16] = o1.w;
    }
}

extern "C" void kernel_launch(void* const* d_in, const int* in_sizes, int n_in,
                              void* d_out, int out_size, void* d_ws, size_t ws_size,
                              hipStream_t stream) {
    const float* fp    = (const float*)d_in[0];   // flow_proportions [B,N,16]
    const float* dem   = (const float*)d_in[1];   // demands [B,N]
    const int*   adj   = (const int*)  d_in[2];   // adj_lst [B,N,16]
    const int*   inidx = (const int*)  d_in[3];   // in_indices [B,N,16,3]
    const int*   nn    = (const int*)  d_in[4];   // num_nodes (device scalar)

    const int TN = in_sizes[1];                   // B*N total nodes
    const size_t E = (size_t)TN * 16;             // edge count

    float* weights = (float*)d_ws;
    int*   gidx    = (int*)  ((char*)d_ws + E * 4);
    float* f0      = (float*)((char*)d_ws + E * 8);
    float* negdem  = (float*)((char*)d_ws + E * 12);
    float* out     = (float*)d_out;               // doubles as ping-pong buffer

    // iteration 1 + all precompute
    prep_kernel<<<(TN + 255) / 256, 256, 0, stream>>>(
        fp, dem, adj, inidx, nn, weights, gidx, f0, negdem, TN);

    const int nwaves = TN / 16;
    const int blocks = (nwaves + 7) / 8;          // 256 threads = 8 waves/block

    // iterations 2..31 (30 launches), ping-pong f0 <-> out in WMMA layout
    float* src = f0;
    float* dst = out;
    for (int it = 0; it < 30; ++it) {
        iter_kernel<<<blocks, 256, 0, stream>>>(src, weights, gidx, negdem, dst, 0, nwaves);
        float* t = src; src = dst; dst = t;
    }
    // iteration 32: src == f0 after an even number of swaps; write standard layout
    iter_kernel<<<blocks, 256, 0, stream>>>(src, weights, gidx, negdem, out, 1, nwaves);
}